// PoseLoss_19799799234747
// MI455X (gfx1250) — compile-verified
//
#include <hip/hip_runtime.h>
#include <math.h>

typedef __attribute__((ext_vector_type(2))) float v2f;
typedef __attribute__((ext_vector_type(8))) float v8f;

#define KS   31
#define RAD  15
#define Bn   32
#define Cn   13
#define Hn   256
#define Wn   256
#define NSLICE (Bn*Cn)          // 416
#define HW   (Hn*Wn)            // 65536

// d_out layout (floats): loss@0, mpe@1, joints@2 (832), thm@834 (27262976),
// pCKh50@27263810, pos_err@27263811 (416)
#define OFF_JOINTS 2
#define OFF_THM    834
#define OFF_PCKH   27263810
#define OFF_PERR   27263811

__global__ __launch_bounds__(256) void pose_heatmap_kernel(
    const float* __restrict__ pred,     // [B*C*H*W]
    const float* __restrict__ targets,  // [B*C*3]
    float* __restrict__ thm_out,        // d_out + OFF_THM
    float* __restrict__ ws_sqsum,       // [416]
    int*   __restrict__ ws_maxi)        // [416]
{
    __shared__ float su0[Hn], su1[Hn], sv0[Wn], sv1[Wn];
    __shared__ float red_sum[8];
    __shared__ float red_maxv[8];
    __shared__ int   red_maxi[8];

    const int wg  = blockIdx.x;        // b*C + c
    const int tid = threadIdx.x;       // 0..255

    const float tx = targets[wg*3 + 0];
    const float ty = targets[wg*3 + 1];
    const int   x0 = (int)tx;
    const int   y0 = (int)ty;
    const float rx = tx - (float)x0;
    const float ry = ty - (float)y0;

    // Reference masks (it compares x against H; H==W==256 here)
    const float m00 = (x0   >= 0 && y0   >= 0 && x0   < Hn && y0   < Hn) ? 1.f : 0.f;
    const float m01 = (x0   >= 0 && y0+1 >= 0 && x0   < Hn && y0+1 < Hn) ? 1.f : 0.f;
    const float m10 = (x0+1 >= 0 && y0   >= 0 && x0+1 < Hn && y0   < Hn) ? 1.f : 0.f;
    const float m11 = (x0+1 >= 0 && y0+1 >= 0 && x0+1 < Hn && y0+1 < Hn) ? 1.f : 0.f;

    // 1-D normalized Gaussian (sigma=2 -> exp(-d^2/8)); normalization constant
    float s = 0.f;
    #pragma unroll
    for (int j = 0; j < KS; ++j) { float d = (float)(j - RAD); s += __expf(-0.125f * d * d); }
    const float invs = 1.0f / s;

    // Build separable LDS tables: thm(y,x) = su0[y]*sv0[x] + su1[y]*sv1[x]
    {
        const int p = tid;                        // covers both y and x (256 each)
        const int dy0 = p - y0, dy1 = p - y0 - 1;
        const float gy0 = (dy0 >= -RAD && dy0 <= RAD) ? __expf(-0.125f*(float)(dy0*dy0))*invs : 0.f;
        const float gy1 = (dy1 >= -RAD && dy1 <= RAD) ? __expf(-0.125f*(float)(dy1*dy1))*invs : 0.f;
        su0[p] = gy0;
        su1[p] = gy1;
        const int dx0 = p - x0, dx1 = p - x0 - 1;
        const float gx0 = (dx0 >= -RAD && dx0 <= RAD) ? __expf(-0.125f*(float)(dx0*dx0))*invs : 0.f;
        const float gx1 = (dx1 >= -RAD && dx1 <= RAD) ? __expf(-0.125f*(float)(dx1*dx1))*invs : 0.f;
        sv0[p] = (1.f - ry) * ((1.f - rx)*m00*gx0 + rx*m10*gx1);
        sv1[p] =        ry  * ((1.f - rx)*m01*gx0 + rx*m11*gx1);
    }
    __syncthreads();

    const int wave = tid >> 5;
    const int lane = tid & 31;
    const int hi   = lane >> 4;        // upper half-wave holds A/B K=2,3 -> zeros
    const int li   = lane & 15;
    const float kmask = (hi == 0) ? 1.f : 0.f;

    const size_t base = (size_t)wg * HW;
    const float* p0 = pred    + base;
    float*       t0 = thm_out + base;

    float acc   = 0.f;
    float bestv = -3.402823466e+38f;
    int   besti = 0x7fffffff;

    // 256 tiles of 16x16 per slice; wave handles col blocks {wave, wave+8},
    // B operand hoisted per col block (reused across 16 row tiles / WMMAs).
    for (int cb = 0; cb < 2; ++cb) {
        const int tx0t = (wave + 8*cb) << 4;

        v2f b;
        b.x = sv0[tx0t + li] * kmask;
        b.y = sv1[tx0t + li] * kmask;

        for (int rb = 0; rb < 16; ++rb) {
            const int ty0t = rb << 4;

            v2f a;
            a.x = su0[ty0t + li] * kmask;
            a.y = su1[ty0t + li] * kmask;

            v8f c = {};
            c = __builtin_amdgcn_wmma_f32_16x16x4_f32(
                    /*neg_a=*/false, a, /*neg_b=*/false, b,
                    /*c_mod=*/(short)0, c, /*reuse_a=*/false, /*reuse_b=*/false);

            // D layout: VGPR r at lane L -> row = r + 8*(L>=16), col = L%16
            const int idx0 = (ty0t + 8*hi) * Wn + tx0t + li;
            const float* pp = p0 + idx0;
            float*       tp = t0 + idx0;

            float pv[8];
            #pragma unroll
            for (int r = 0; r < 8; ++r) pv[r] = pp[r * Wn];   // 8 loads in flight

            #pragma unroll
            for (int r = 0; r < 8; ++r) {
                const float hm = c[r];
                const float d  = hm - pv[r];
                acc = fmaf(d, d, acc);
                tp[r * Wn] = hm;
                const int idx = idx0 + r * Wn;
                // branchless first-index argmax (v_cmp + v_cndmask, no EXEC games)
                const bool better = (pv[r] > bestv) | ((pv[r] == bestv) & (idx < besti));
                bestv = better ? pv[r] : bestv;
                besti = better ? idx   : besti;
            }
        }
    }

    // intra-wave reduction (wave32)
    #pragma unroll
    for (int off = 16; off > 0; off >>= 1) {
        acc += __shfl_xor(acc, off, 32);
        const float ov = __shfl_xor(bestv, off, 32);
        const int   oi = __shfl_xor(besti, off, 32);
        const bool better = (ov > bestv) | ((ov == bestv) & (oi < besti));
        bestv = better ? ov : bestv;
        besti = better ? oi : besti;
    }
    if (lane == 0) { red_sum[wave] = acc; red_maxv[wave] = bestv; red_maxi[wave] = besti; }
    __syncthreads();

    if (tid == 0) {
        float s2 = 0.f;
        float bv = red_maxv[0];
        int   bi = red_maxi[0];
        #pragma unroll
        for (int wv = 0; wv < 8; ++wv) {
            s2 += red_sum[wv];
            const float ov = red_maxv[wv];
            const int   oi = red_maxi[wv];
            const bool better = (ov > bv) | ((ov == bv) & (oi < bi));
            bv = better ? ov : bv;
            bi = better ? oi : bi;
        }
        ws_sqsum[wg] = s2;
        ws_maxi[wg]  = bi;
    }
}

__global__ __launch_bounds__(512) void pose_finalize_kernel(
    const float* __restrict__ targets,
    const float* __restrict__ head_size,
    const float* __restrict__ ws_sqsum,
    const int*   __restrict__ ws_maxi,
    float* __restrict__ out)
{
    __shared__ float s_ev[512], s_vis[512], s_inl[512], s_sq[512];
    const int i = threadIdx.x;

    float ev = 0.f, vis = 0.f, inl = 0.f, sq = 0.f;
    if (i < NSLICE) {
        const int idx = ws_maxi[i];
        const float xf = (float)(idx & (Wn - 1));
        const float yf = floorf((float)idx / (float)Wn);
        out[OFF_JOINTS + i*2 + 0] = xf;
        out[OFF_JOINTS + i*2 + 1] = yf;

        const float tx = targets[i*3 + 0];
        const float ty = targets[i*3 + 1];
        const float v  = targets[i*3 + 2];
        const float dx = xf - tx, dy = yf - ty;
        const float err = sqrtf(dx*dx + dy*dy);
        out[OFF_PERR + i] = err;

        const int b = i / Cn;
        const float thr = head_size[b] * 0.5f;
        ev  = err * v;
        vis = v;
        inl = (err <= thr) ? v : 0.f;
        sq  = ws_sqsum[i];
    }
    s_ev[i] = ev; s_vis[i] = vis; s_inl[i] = inl; s_sq[i] = sq;
    __syncthreads();

    for (int off = 256; off > 0; off >>= 1) {
        if (i < off) {
            s_ev[i]  += s_ev[i + off];
            s_vis[i] += s_vis[i + off];
            s_inl[i] += s_inl[i + off];
            s_sq[i]  += s_sq[i + off];
        }
        __syncthreads();
    }

    if (i == 0) {
        const float eps = 0.001f;
        out[0]        = s_sq[0] / (float)NSLICE;           // loss = mean over (b,c) of per-slice SSE
        out[1]        = s_ev[0] / (eps + s_vis[0]);        // mean_position_error_2d
        out[OFF_PCKH] = s_inl[0] / (eps + s_vis[0]);       // pCKh50
    }
}

extern "C" void kernel_launch(void* const* d_in, const int* in_sizes, int n_in,
                              void* d_out, int out_size, void* d_ws, size_t ws_size,
                              hipStream_t stream) {
    (void)in_sizes; (void)n_in; (void)out_size; (void)ws_size;

    const float* pred    = (const float*)d_in[0];
    const float* targets = (const float*)d_in[1];
    const float* head    = (const float*)d_in[2];
    float* out = (float*)d_out;

    float* ws_sqsum = (float*)d_ws;          // 512 floats
    int*   ws_maxi  = (int*)(ws_sqsum + 512);

    pose_heatmap_kernel<<<NSLICE, 256, 0, stream>>>(
        pred, targets, out + OFF_THM, ws_sqsum, ws_maxi);

    pose_finalize_kernel<<<1, 512, 0, stream>>>(
        targets, head, ws_sqsum, ws_maxi, out);
}